// Decoder_35450660061950
// MI455X (gfx1250) — compile-verified
//
#include <hip/hip_runtime.h>
#include <hip/hip_bf16.h>

#define NEG_SLOPE 0.2f

typedef __attribute__((ext_vector_type(2))) float v2f;
typedef __attribute__((ext_vector_type(8))) float v8f;

// ---------------------------------------------------------------------------
// f32 GEMM on the WMMA pipe: C[M,F] = A[M,K] @ B[K,F], one wave per 16x16 tile.
// V_WMMA_F32_16X16X4_F32 fragments per the CDNA5 ISA lane layouts:
//   A 16x4:  lanes 0-15 -> M=lane, VGPR{0,1}=K{0,1}; lanes 16-31 -> K{2,3}
//   B 4x16:  lanes 0-15 -> N=lane, VGPR{0,1}=K{0,1}; lanes 16-31 -> K{2,3}
//   C 16x16: VGPR r: lanes 0-15 -> M=r, lanes 16-31 -> M=r+8; N=lane&15
// Bounds handling is branchless: clamp addresses in-range, mask with selects,
// so the inner loop is 4 unconditional b32 loads + 4 v_cndmask + 1 v_wmma.
// ---------------------------------------------------------------------------
__global__ __launch_bounds__(256)
void gat_wmma_gemm_f32(const float* __restrict__ A, const float* __restrict__ B,
                       float* __restrict__ C, int M, int K, int F) {
    const int wavesPerBlock = blockDim.x >> 5;
    const int wid  = blockIdx.x * wavesPerBlock + (threadIdx.x >> 5);
    const int lane = threadIdx.x & 31;
    const int nTiles = (F + 15) >> 4;
    const int mTiles = (M + 15) >> 4;
    if (wid >= mTiles * nTiles) return;            // wave-uniform exit (EXEC all-1 at WMMA)

    const int mt = wid / nTiles;
    const int nt = wid - mt * nTiles;
    const int m0 = mt << 4, n0 = nt << 4;
    const int laneHi = lane >> 4;                  // 0|1
    const int laneLo = lane & 15;

    const int  rowA  = m0 + laneLo;
    const bool rowOK = rowA < M;
    const size_t rA  = (size_t)min(rowA, M - 1) * (size_t)K;

    const int  colB  = n0 + laneLo;
    const bool colOK = colB < F;
    const int  cB    = min(colB, F - 1);

    v8f c = {0.f,0.f,0.f,0.f,0.f,0.f,0.f,0.f};

    for (int k0 = 0; k0 < K; k0 += 4) {
        const int  ka  = k0 + laneHi * 2;
        const bool k0OK = ka < K;
        const bool k1OK = (ka + 1) < K;
        const int  kc0 = min(ka,     K - 1);
        const int  kc1 = min(ka + 1, K - 1);

        const float ax = A[rA + kc0];
        const float ay = A[rA + kc1];
        const float bx = B[(size_t)kc0 * F + cB];
        const float by = B[(size_t)kc1 * F + cB];

        v2f a, b;
        a.x = (rowOK & k0OK) ? ax : 0.f;
        a.y = (rowOK & k1OK) ? ay : 0.f;
        b.x = (colOK & k0OK) ? bx : 0.f;
        b.y = (colOK & k1OK) ? by : 0.f;

        c = __builtin_amdgcn_wmma_f32_16x16x4_f32(false, a, false, b,
                                                  (short)0, c, false, false);
    }

#pragma unroll
    for (int r = 0; r < 8; ++r) {
        const int row = m0 + r + laneHi * 8;
        if (row < M && colOK) C[(size_t)row * F + colB] = c[r];
    }
}

// ---------------------------------------------------------------------------
// Per-node attention dots: esrc[i] = h[i,:].a_src, edst[i] = h[i,:].a_dst
// ---------------------------------------------------------------------------
template<int CF>
__global__ void gat_node_dots_t(const float* __restrict__ h,
                                const float* __restrict__ asrc,
                                const float* __restrict__ adst,
                                float* __restrict__ es, float* __restrict__ ed,
                                int N, int Frt) {
    const int F = CF ? CF : Frt;
    const int i = blockIdx.x * blockDim.x + threadIdx.x;
    if (i >= N) return;
    float s = 0.f, d = 0.f;
    const float* row = h + (size_t)i * F;
    for (int f = 0; f < F; ++f) { const float v = row[f]; s += v * asrc[f]; d += v * adst[f]; }
    es[i] = s; ed[i] = d;
}

// Init: segment-max buffer to -inf, denom to 0
__global__ void gat_fill_seg(float* __restrict__ m, float* __restrict__ den, int N) {
    const int i = blockIdx.x * blockDim.x + threadIdx.x;
    if (i >= N) return;
    m[i] = -__builtin_inff();
    den[i] = 0.f;
}

__device__ __forceinline__ void atomicMaxF32(float* addr, float val) {
    if (val >= 0.f) atomicMax((int*)addr, __float_as_int(val));
    else            atomicMin((unsigned int*)addr, __float_as_uint(val));
}

__device__ __forceinline__ void edge_endpoints(const int* __restrict__ ei,
                                               long long e, long long E,
                                               int& s, int& d) {
    if (e < E) { s = ei[e]; d = ei[E + e]; }
    else       { s = d = (int)(e - E); }            // appended self-loops
}

__device__ __forceinline__ int edge_dst(const int* __restrict__ ei,
                                        long long e, long long E) {
    return (e < E) ? ei[E + e] : (int)(e - E);
}

// Pass 1: e = leaky_relu(es[src]+ed[dst]); segment max over dst
__global__ void gat_edge_logits(const float* __restrict__ es, const float* __restrict__ ed,
                                const int* __restrict__ ei, float* __restrict__ ebuf,
                                float* __restrict__ mbuf, long long E, long long Etot) {
    const long long e = (long long)blockIdx.x * blockDim.x + threadIdx.x;
    if (e >= Etot) return;
    int s, d; edge_endpoints(ei, e, E, s, d);
    const float v = es[s] + ed[d];
    const float l = v > 0.f ? v : NEG_SLOPE * v;
    ebuf[e] = l;
    atomicMaxF32(&mbuf[d], l);
}

// Pass 2: ex = exp(e - m[dst]) (in place); segment sum over dst
__global__ void gat_edge_exp(float* __restrict__ ebuf, const float* __restrict__ mbuf,
                             float* __restrict__ den, const int* __restrict__ ei,
                             long long E, long long Etot) {
    const long long e = (long long)blockIdx.x * blockDim.x + threadIdx.x;
    if (e >= Etot) return;
    const int d = edge_dst(ei, e, E);
    const float x = __expf(ebuf[e] - mbuf[d]);
    ebuf[e] = x;
    atomicAdd(&den[d], x);
}

// Pass 3: out[dst,f] += (ex/denom[dst]) * h[src,f]. Consecutive lanes take
// consecutive f of the same edge -> coalesced h reads + coalesced atomics.
// CF>0 makes the t/F division a compile-time-constant strength reduction.
template<int CF>
__global__ void gat_edge_aggregate_t(const float* __restrict__ ebuf, const float* __restrict__ den,
                                     const float* __restrict__ h, const int* __restrict__ ei,
                                     float* __restrict__ out, long long E, long long Etot, int Frt) {
    const int F = CF ? CF : Frt;
    const long long t = (long long)blockIdx.x * blockDim.x + threadIdx.x;
    const long long total = Etot * (long long)F;
    if (t >= total) return;
    const long long e = t / F;
    const int f = (int)(t - e * F);
    int s, d; edge_endpoints(ei, e, E, s, d);
    const float alpha = ebuf[e] / den[d];
    atomicAdd(&out[(size_t)d * F + f], alpha * h[(size_t)s * F + f]);
}

// out = relu(out + b) in place
template<int CF>
__global__ void gat_bias_relu_t(float* __restrict__ out, const float* __restrict__ b,
                                long long total, int Frt) {
    const int F = CF ? CF : Frt;
    const long long t = (long long)blockIdx.x * blockDim.x + threadIdx.x;
    if (t >= total) return;
    const int f = (int)(t % F);
    const float v = out[t] + b[f];
    out[t] = v > 0.f ? v : 0.f;
}

// ---------------------------------------------------------------------------
static inline int cdiv_i(long long a, long long b) { return (int)((a + b - 1) / b); }

static void run_gat_layer(const float* hin, int Fin, const float* W, int Fout,
                          const float* asrc, const float* adst, const float* bias,
                          const int* ei, long long E, int N,
                          float* htr,           // [N,Fout] transform scratch
                          float* ebuf, float* mbuf, float* den,
                          float* es, float* ed,
                          float* out,           // [N,Fout], pre-zeroed
                          hipStream_t stream) {
    const long long Etot = E + N;

    // Pick constant-F instantiations for the hot kernels.
    auto* dots = (Fout == 50)  ? gat_node_dots_t<50>  :
                 (Fout == 100) ? gat_node_dots_t<100> : gat_node_dots_t<0>;
    auto* agg  = (Fout == 50)  ? gat_edge_aggregate_t<50>  :
                 (Fout == 100) ? gat_edge_aggregate_t<100> : gat_edge_aggregate_t<0>;
    auto* brel = (Fout == 50)  ? gat_bias_relu_t<50>  :
                 (Fout == 100) ? gat_bias_relu_t<100> : gat_bias_relu_t<0>;

    // Transform on the WMMA pipe.
    {
        const int mT = (N + 15) / 16, nT = (Fout + 15) / 16;
        const int waves = mT * nT;
        gat_wmma_gemm_f32<<<cdiv_i(waves, 8), 256, 0, stream>>>(hin, W, htr, N, Fin, Fout);
    }
    dots<<<cdiv_i(N, 256), 256, 0, stream>>>(htr, asrc, adst, es, ed, N, Fout);
    gat_fill_seg<<<cdiv_i(N, 256), 256, 0, stream>>>(mbuf, den, N);
    gat_edge_logits<<<cdiv_i(Etot, 256), 256, 0, stream>>>(es, ed, ei, ebuf, mbuf, E, Etot);
    gat_edge_exp<<<cdiv_i(Etot, 256), 256, 0, stream>>>(ebuf, mbuf, den, ei, E, Etot);
    agg<<<cdiv_i(Etot * Fout, 256), 256, 0, stream>>>(ebuf, den, htr, ei, out, E, Etot, Fout);
    brel<<<cdiv_i((long long)N * Fout, 256), 256, 0, stream>>>(out, bias,
                                                               (long long)N * Fout, Fout);
}

extern "C" void kernel_launch(void* const* d_in, const int* in_sizes, int n_in,
                              void* d_out, int out_size, void* d_ws, size_t ws_size,
                              hipStream_t stream) {
    const float* x     = (const float*)d_in[0];
    const int*   ei    = (const int*)d_in[1];     // edge_index [2,E] flat
    const float* W1    = (const float*)d_in[2];
    const float* asrc1 = (const float*)d_in[3];
    const float* adst1 = (const float*)d_in[4];
    const float* b1    = (const float*)d_in[5];
    const float* W2    = (const float*)d_in[6];
    const float* asrc2 = (const float*)d_in[7];
    const float* adst2 = (const float*)d_in[8];
    const float* b2    = (const float*)d_in[9];

    const int Fin = 4;
    const int N   = in_sizes[0] / Fin;            // 100000
    const int F1  = in_sizes[2] / Fin;            // 50
    const int F2  = in_sizes[6] / F1;             // 100
    const long long E    = (long long)in_sizes[1] / 2;
    const long long Etot = E + N;

    float* ws   = (float*)d_ws;
    float* h1   = ws;                             // [N,F1] transform
    float* out1 = h1   + (size_t)N * F1;          // [N,F1] layer-1 output -> layer-2 input
    float* h2   = out1 + (size_t)N * F1;          // [N,F2] transform
    float* ebuf = h2   + (size_t)N * F2;          // [Etot]
    float* mbuf = ebuf + Etot;                    // [N]
    float* den  = mbuf + N;                       // [N]
    float* es   = den  + N;                       // [N]
    float* ed   = es   + N;                       // [N]

    float* outF = (float*)d_out;                  // [N,F2] then edge_index copy

    hipMemsetAsync(out1, 0, (size_t)N * F1 * sizeof(float), stream);
    hipMemsetAsync(outF, 0, (size_t)N * F2 * sizeof(float), stream);

    // Layer 1: x[N,4] -> out1[N,50]
    run_gat_layer(x, Fin, W1, F1, asrc1, adst1, b1, ei, E, N,
                  h1, ebuf, mbuf, den, es, ed, out1, stream);
    // Layer 2: out1[N,50] -> outF[N,100]
    run_gat_layer(out1, F1, W2, F2, asrc2, adst2, b2, ei, E, N,
                  h2, ebuf, mbuf, den, es, ed, outF, stream);

    // Tuple output: append edge_index after the [N,F2] feature block.
    hipMemcpyAsync(outF + (size_t)N * F2, d_in[1],
                   (size_t)in_sizes[1] * sizeof(int),
                   hipMemcpyDeviceToDevice, stream);
}